// L_MF_32040456028890
// MI455X (gfx1250) — compile-verified
//
#include <hip/hip_runtime.h>

// Problem constants from setup_inputs(): org is [B=16, C=3, H=512, W=512] fp32.
#define HH   512
#define WW   512
#define CCH  3
#define TW   64            // tile width  (output pixels)
#define TH   16            // tile height (output pixels)
#define HROWS (TH + 2)     // 18 halo rows
#define HCOLS (TW + 2)     // 66 halo cols
#define HS   72            // padded LDS row stride in floats (288B, 16B multiple)
#define NTHREADS 256

#if defined(__HIP_DEVICE_COMPILE__) && __has_builtin(__builtin_amdgcn_global_load_async_to_lds_b32)
#define USE_ASYNC_LDS 1
#else
#define USE_ASYNC_LDS 0
#endif

typedef __attribute__((address_space(1))) int g_int;
typedef __attribute__((address_space(3))) int l_int;

__device__ __forceinline__ float min3f(float a, float b, float c) { return fminf(fminf(a, b), c); }
__device__ __forceinline__ float max3f(float a, float b, float c) { return fmaxf(fmaxf(a, b), c); }
__device__ __forceinline__ float med3f(float a, float b, float c) { return __builtin_amdgcn_fmed3f(a, b, c); }

// Load 6 consecutive floats from 16B-aligned LDS address: b128 + b64.
__device__ __forceinline__ void load6(const float* p, float* r) {
    float4 v4 = *reinterpret_cast<const float4*>(p);
    float2 v2 = *reinterpret_cast<const float2*>(p + 4);
    r[0] = v4.x; r[1] = v4.y; r[2] = v4.z; r[3] = v4.w; r[4] = v2.x; r[5] = v2.y;
}

__global__ __launch_bounds__(NTHREADS)
void median_shift_energy_kernel(const float* __restrict__ org, float* __restrict__ out) {
    __shared__ __align__(16) float smem[CCH][HROWS][HS];

    const int b  = blockIdx.z;
    const int y0 = blockIdx.y * TH;
    const int x0 = blockIdx.x * TW;
    const float* gsrc = org + (size_t)b * CCH * HH * WW;

    // Block-uniform predicate: no divergence, scalar branch.
    const bool interior = (y0 > 0) && (y0 + TH < HH) && (x0 > 0) && (x0 + TW < WW);

    // ---------------- stage 3-channel halo tile into LDS ----------------
#if USE_ASYNC_LDS
    if (interior) {
        // Fast path (70% of blocks): unguarded async DMA global->LDS.
        for (int idx = threadIdx.x; idx < CCH * HROWS * HCOLS; idx += NTHREADS) {
            const int c   = idx / (HROWS * HCOLS);
            const int rem = idx - c * (HROWS * HCOLS);
            const int r   = rem / HCOLS;
            const int col = rem - r * HCOLS;
            const float* src = gsrc + ((size_t)c * HH + (y0 - 1 + r)) * WW + (x0 - 1 + col);
            __builtin_amdgcn_global_load_async_to_lds_b32(
                (g_int*)src, (l_int*)&smem[c][r][col], 0, 0);
        }
    } else {
        // Border path: pre-zero tile (zero padding), then DMA in-bounds lanes.
        for (int idx = threadIdx.x; idx < CCH * HROWS * HS; idx += NTHREADS)
            ((float*)smem)[idx] = 0.0f;
        __syncthreads();   // ds stores complete before async engine writes

        for (int idx = threadIdx.x; idx < CCH * HROWS * HCOLS; idx += NTHREADS) {
            const int c   = idx / (HROWS * HCOLS);
            const int rem = idx - c * (HROWS * HCOLS);
            const int r   = rem / HCOLS;
            const int col = rem - r * HCOLS;
            const int gy  = y0 - 1 + r;
            const int gx  = x0 - 1 + col;
            if ((unsigned)gy < (unsigned)HH && (unsigned)gx < (unsigned)WW) {
                const float* src = gsrc + ((size_t)c * HH + gy) * WW + gx;
                __builtin_amdgcn_global_load_async_to_lds_b32(
                    (g_int*)src, (l_int*)&smem[c][r][col], 0, 0);
            }
        }
    }
#if __has_builtin(__builtin_amdgcn_s_wait_asynccnt)
    __builtin_amdgcn_s_wait_asynccnt(0);
#else
    asm volatile("s_wait_asynccnt 0" ::: "memory");
#endif
    __syncthreads();
#else
    // Fallback: coalesced global loads -> LDS stores, zero for out-of-image.
    for (int idx = threadIdx.x; idx < CCH * HROWS * HCOLS; idx += NTHREADS) {
        const int c   = idx / (HROWS * HCOLS);
        const int rem = idx - c * (HROWS * HCOLS);
        const int r   = rem / HCOLS;
        const int col = rem - r * HCOLS;
        const int gy  = y0 - 1 + r;
        const int gx  = x0 - 1 + col;
        float v = 0.0f;
        if ((unsigned)gy < (unsigned)HH && (unsigned)gx < (unsigned)WW)
            v = gsrc[((size_t)c * HH + gy) * WW + gx];
        smem[c][r][col] = v;
    }
    __syncthreads();
#endif

    // ---------------- compute: 4 adjacent pixels per thread ----------------
    const int lx = (threadIdx.x & 15) << 2;  // 0,4,...,60
    const int ly = threadIdx.x >> 4;         // 0..15

    float acc[4] = {0.f, 0.f, 0.f, 0.f};

#pragma unroll
    for (int c = 0; c < CCH; ++c) {
        float r0[6], r1[6], r2[6];
        load6(&smem[c][ly + 0][lx], r0);   // dy = -1 row
        load6(&smem[c][ly + 1][lx], r1);   // dy =  0 row (center)
        load6(&smem[c][ly + 2][lx], r2);   // dy = +1 row

#pragma unroll
        for (int i = 0; i < 4; ++i) {
            // 9 samples (reference's duplicated (+1,+1), missing (+1,0)):
            //   row-1: r0[i], r0[i+1], r0[i+2]
            //   row 0: r1[i], r1[i+1], r1[i+2]
            //   row+1: r2[i], r2[i+2], r2[i+2]
            const float mn0 = min3f(r0[i], r0[i + 1], r0[i + 2]);
            const float md0 = med3f(r0[i], r0[i + 1], r0[i + 2]);
            const float mx0 = max3f(r0[i], r0[i + 1], r0[i + 2]);
            const float mn1 = min3f(r1[i], r1[i + 1], r1[i + 2]);
            const float md1 = med3f(r1[i], r1[i + 1], r1[i + 2]);
            const float mx1 = max3f(r1[i], r1[i + 1], r1[i + 2]);
            // trio (a, b, b) sorts as (min(a,b), b, max(a,b))
            const float mn2 = fminf(r2[i], r2[i + 2]);
            const float md2 = r2[i + 2];
            const float mx2 = fmaxf(r2[i], r2[i + 2]);

            const float t0  = max3f(mn0, mn1, mn2);
            const float t1  = med3f(md0, md1, md2);
            const float t2  = min3f(mx0, mx1, mx2);
            const float med = med3f(t0, t1, t2);

            const float d = med - r1[i + 1];   // center sample == org value
            acc[i] = fmaf(d, d, acc[i]);
        }
    }

    float4 o;
    o.x = acc[0]; o.y = acc[1]; o.z = acc[2]; o.w = acc[3];
    float* dst = out + ((size_t)b * HH + (y0 + ly)) * WW + (x0 + lx);
    *reinterpret_cast<float4*>(dst) = o;
}

extern "C" void kernel_launch(void* const* d_in, const int* in_sizes, int n_in,
                              void* d_out, int out_size, void* d_ws, size_t ws_size,
                              hipStream_t stream) {
    const float* org = (const float*)d_in[0];
    float* out = (float*)d_out;
    const int B = in_sizes[0] / (CCH * HH * WW);   // 16
    dim3 grid(WW / TW, HH / TH, B);                // 8 x 32 x 16 = 4096 blocks
    median_shift_energy_kernel<<<grid, dim3(NTHREADS), 0, stream>>>(org, out);
}